// CPIPre_27659589386637
// MI455X (gfx1250) — compile-verified
//
#include <hip/hip_runtime.h>
#include <hip/hip_bf16.h>

typedef float v2f __attribute__((ext_vector_type(2)));
typedef float v8f __attribute__((ext_vector_type(8)));
typedef unsigned int v4u __attribute__((ext_vector_type(4)));
typedef int v4i __attribute__((ext_vector_type(4)));
typedef int v8i __attribute__((ext_vector_type(8)));

#define N_ATOMS 8192
#define D_EMB   10
#define DP      16          // padded feature dim for WMMA tiles
#define L_WORDS 4096
#define KWIN    11
#define ROWS    128         // output rows per workgroup in GEMM (8 waves x 16)
#define KC      64          // K chunk per double-buffer stage
#define APITCH  68          // LDS row pitch (floats): 64 data + 4 pad -> conflict-free
#define KSPLIT  16          // split-K factor
#define KRANGE  (N_ATOMS / KSPLIT)   // 512

#if __has_builtin(__builtin_amdgcn_tensor_load_to_lds)
#define HAVE_TDM 1
#else
#define HAVE_TDM 0
#endif

// ---------------- GNN: gather fingerprint embeddings into padded xs ----------------
__global__ void k_gather_fp(const int* __restrict__ fp, const float* __restrict__ emb,
                            float* __restrict__ xs) {
  int t = blockIdx.x * blockDim.x + threadIdx.x;   // over N_ATOMS*DP
  int i = t >> 4, d = t & 15;
  xs[t] = (d < D_EMB) ? emb[fp[i] * D_EMB + d] : 0.0f;
}

// hs = relu(xs @ W^T + b), written padded to 16 cols (cols 10..15 = 0)
__global__ void k_gnn_hidden(const float* __restrict__ xs, const float* __restrict__ W,
                             const float* __restrict__ b, float* __restrict__ hs) {
  int i = blockIdx.x * blockDim.x + threadIdx.x;   // N_ATOMS rows
  float x[D_EMB];
#pragma unroll
  for (int d = 0; d < D_EMB; ++d) x[d] = xs[i * DP + d];
#pragma unroll
  for (int o = 0; o < D_EMB; ++o) {
    float acc = b[o];
#pragma unroll
    for (int d = 0; d < D_EMB; ++d) acc = fmaf(W[o * D_EMB + d], x[d], acc);
    hs[i * DP + o] = fmaxf(acc, 0.0f);
  }
#pragma unroll
  for (int o = D_EMB; o < DP; ++o) hs[i * DP + o] = 0.0f;
}

__global__ void k_copy4(const float4* __restrict__ src, float4* __restrict__ dst) {
  int t = blockIdx.x * blockDim.x + threadIdx.x;
  dst[t] = src[t];
}

#if HAVE_TDM
// Issue one TDM 2D tile load: tileW x tileH elements (f32), row stride strideElts,
// into LDS at byte address ldsByte. pad=true inserts 4 dwords after every 64 dwords
// (row pitch 68 floats -> matches APITCH, bank-conflict-free WMMA A reads).
__device__ __forceinline__ void tdm_load_tile(const float* gsrc, unsigned ldsByte,
                                              unsigned tileW, unsigned tileH,
                                              unsigned long long strideElts, bool pad) {
  unsigned long long ga = (unsigned long long)(size_t)gsrc;
  v4u g0;
  g0[0] = 1u;                                               // count=1, user mode
  g0[1] = ldsByte;                                          // lds_addr
  g0[2] = (unsigned)ga;                                     // global_addr[31:0]
  g0[3] = (unsigned)((ga >> 32) & 0x01FFFFFFu) | (2u << 30);// global_addr[56:32], type=2
  const unsigned td0 = 1u << 24;   // huge tensor dims: no OOB clipping
  const unsigned td1 = 1u << 24;
  unsigned dw0 = (2u << 16);                                // data_size = 4 bytes
  if (pad) dw0 |= (1u << 20) | (5u << 22) | (3u << 25);     // pad: every 64 dw, +4 dw
  v8i g1;
  g1[0] = (int)dw0;
  g1[1] = (int)((td0 & 0xffffu) << 16);                     // tensor_dim0[15:0]
  g1[2] = (int)(((td1 & 0xffffu) << 16) | (td0 >> 16));     // td1[15:0] | td0[31:16]
  g1[3] = (int)((tileW << 16) | (td1 >> 16));               // tile_dim0 | td1[31:16]
  g1[4] = (int)tileH;                                       // tile_dim1 (tile_dim2=0)
  g1[5] = (int)(unsigned)strideElts;                        // dim0 stride [31:0]
  g1[6] = (int)(unsigned)((strideElts >> 32) & 0xffffu);    // dim0 stride [47:32]
  g1[7] = 0;
  v4i gz4 = {0, 0, 0, 0};
#if defined(__clang_major__) && (__clang_major__ >= 23)
  v8i gz8 = {0, 0, 0, 0, 0, 0, 0, 0};
  __builtin_amdgcn_tensor_load_to_lds(g0, g1, gz4, gz4, gz8, 0);
#else
  __builtin_amdgcn_tensor_load_to_lds(g0, g1, gz4, gz4, 0);
#endif
}
#endif

// y += adjacency @ hsPad  (y pre-initialized with xs; split-K atomic accumulate)
// grid = (N_ATOMS/ROWS) * KSPLIT blocks of 256 threads (8 waves, 16 rows each)
__global__ void k_adj_gemm(const float* __restrict__ adj, const float* __restrict__ hsPad,
                           float* __restrict__ yPad) {
#if HAVE_TDM
  __shared__ float aT[2][ROWS * APITCH];   // 2 x 34816 B, TDM double buffer
  __shared__ float bAll[KRANGE * DP];      // 32 KB: whole K-range of hsPad for this split

  const int wg = blockIdx.x;
  const int rowBlk = wg / KSPLIT;
  const int kBlk   = wg % KSPLIT;
  const int row0   = rowBlk * ROWS;
  const size_t k0base = (size_t)kBlk * KRANGE;

  const int tid  = threadIdx.x;
  const int lane = tid & 31;
  const int half = lane >> 4;
  const int n    = lane & 15;
  const int wave = __builtin_amdgcn_readfirstlane(tid) >> 5;   // SGPR-uniform wave id
  const int myRow = wave * 16 + n;

  // ---- one-time cooperative load of B: hsPad rows [k0base, k0base+KRANGE) ----
#pragma unroll
  for (int s = 0; s < 8; ++s) {
    int f4 = tid + s * 256;                 // float4 index over KRANGE*DP/4 = 2048
    *(float4*)(&bAll[f4 * 4]) = *(const float4*)(hsPad + k0base * DP + f4 * 4);
  }

  // ---- prime TDM pipeline: each wave DMAs its own 16x64 slab (no barriers) ----
  const unsigned aBase = (unsigned)(size_t)&aT[0][0];
  const unsigned slabOff = (unsigned)(wave * 16 * APITCH) * 4u;
  const unsigned bufStride = (unsigned)(ROWS * APITCH) * 4u;
  const float* gslab = adj + (size_t)(row0 + wave * 16) * N_ATOMS + k0base;

  tdm_load_tile(gslab, aBase + slabOff, KC, 16, N_ATOMS, true);

  __syncthreads();   // bAll visible to all waves (also covers nothing for aT: per-wave)

  v8f c = {0.f, 0.f, 0.f, 0.f, 0.f, 0.f, 0.f, 0.f};
  int buf = 0;
  for (int kc = 0; kc < KRANGE; kc += KC) {
    if (kc + KC < KRANGE) {
      tdm_load_tile(gslab + kc + KC, aBase + (buf ^ 1) * bufStride + slabOff,
                    KC, 16, N_ATOMS, true);
      __builtin_amdgcn_s_wait_tensorcnt(1);   // current buffer's DMA done
    } else {
      __builtin_amdgcn_s_wait_tensorcnt(0);
    }
    const float* aSlab = &aT[buf][0];
#pragma unroll
    for (int kk = 0; kk < KC; kk += 4) {
      v2f a = *(const v2f*)(&aSlab[myRow * APITCH + kk + 2 * half]);
      v2f b;
      b.x = bAll[(kc + kk + 2 * half) * DP + n];
      b.y = bAll[(kc + kk + 2 * half + 1) * DP + n];
      c = __builtin_amdgcn_wmma_f32_16x16x4_f32(false, a, false, b, (short)0, c,
                                                false, false);
    }
    buf ^= 1;
  }
#pragma unroll
  for (int v = 0; v < 8; ++v) {
    int m = v + 8 * half;
    atomicAdd(&yPad[(size_t)(row0 + wave * 16 + m) * DP + n], c[v]);
  }
#else  // ------------------- fallback: synchronous LDS staging -------------------
  __shared__ float aTile[ROWS * APITCH];
  __shared__ float bTile[KC * DP];
  const int wg = blockIdx.x;
  const int rowBlk = wg / KSPLIT;
  const int kBlk   = wg % KSPLIT;
  const int row0   = rowBlk * ROWS;
  const size_t k0base = (size_t)kBlk * KRANGE;
  const int tid  = threadIdx.x;
  const int wave = tid >> 5;
  const int lane = tid & 31;
  const int half = lane >> 4;
  const int n    = lane & 15;
  const int myRow = wave * 16 + n;
  v8f c = {0.f, 0.f, 0.f, 0.f, 0.f, 0.f, 0.f, 0.f};
  for (int kc = 0; kc < KRANGE; kc += KC) {
    const size_t k0 = k0base + kc;
#pragma unroll
    for (int s = 0; s < 8; ++s) {
      int f4 = tid + s * 256;
      int r  = f4 >> 4;
      int cq = (f4 & 15) << 2;
      float4 v = *(const float4*)(adj + (size_t)(row0 + r) * N_ATOMS + k0 + cq);
      *(float4*)(&aTile[r * APITCH + cq]) = v;
    }
    {
      int r  = tid >> 2;
      int cq = (tid & 3) << 2;
      *(float4*)(&bTile[r * DP + cq]) = *(const float4*)(hsPad + (k0 + r) * DP + cq);
    }
    __syncthreads();
#pragma unroll
    for (int kk = 0; kk < KC; kk += 4) {
      v2f a = *(const v2f*)(&aTile[myRow * APITCH + kk + 2 * half]);
      v2f b;
      b.x = bTile[(kk + 2 * half) * DP + n];
      b.y = bTile[(kk + 2 * half + 1) * DP + n];
      c = __builtin_amdgcn_wmma_f32_16x16x4_f32(false, a, false, b, (short)0, c,
                                                false, false);
    }
    __syncthreads();
  }
#pragma unroll
  for (int v = 0; v < 8; ++v) {
    int m = v + 8 * half;
    atomicAdd(&yPad[(size_t)(row0 + wave * 16 + m) * DP + n], c[v]);
  }
#endif
}

// compound[d] = mean over rows of xsPad (padded cols are zero)
__global__ void k_mean_rows(const float* __restrict__ xs, float* __restrict__ outv) {
  __shared__ float red[256];
  int t = threadIdx.x, d = t & 15, g = t >> 4;
  float s = 0.f;
  for (int r = g; r < N_ATOMS; r += 16) s += xs[r * DP + d];
  red[t] = s;
  __syncthreads();
  if (t < 16) {
    float acc = 0.f;
    for (int g2 = 0; g2 < 16; ++g2) acc += red[g2 * 16 + t];
    outv[t] = acc / (float)N_ATOMS;
  }
}

// ---------------- Protein branch ----------------
__global__ void k_gather_word(const int* __restrict__ wd, const float* __restrict__ emb,
                              float* __restrict__ ws) {
  int t = blockIdx.x * blockDim.x + threadIdx.x;   // L_WORDS*D_EMB
  int l = t / D_EMB, d = t % D_EMB;
  ws[t] = emb[wd[l] * D_EMB + d];
}

__global__ void k_conv(const float* __restrict__ in, const float* __restrict__ w,
                       const float* __restrict__ bptr, int layer, float* __restrict__ out) {
  int t = blockIdx.x * blockDim.x + threadIdx.x;
  int l = t / D_EMB, d = t % D_EMB;
  const float* wk = w + layer * KWIN * KWIN;
  float acc = bptr[layer];
#pragma unroll
  for (int a = 0; a < KWIN; ++a) {
    int ll = l + a - 5;
    if ((unsigned)ll < (unsigned)L_WORDS) {
#pragma unroll
      for (int b2 = 0; b2 < KWIN; ++b2) {
        int dd = d + b2 - 5;
        if ((unsigned)dd < (unsigned)D_EMB)
          acc = fmaf(wk[a * KWIN + b2], in[ll * D_EMB + dd], acc);
      }
    }
  }
  out[t] = fmaxf(acc, 0.0f);
}

__global__ void k_att(const float* __restrict__ ws, const float* __restrict__ Wa,
                      const float* __restrict__ ba, float* __restrict__ hsAtt) {
  int l = blockIdx.x * blockDim.x + threadIdx.x;   // L_WORDS
  float x[D_EMB];
#pragma unroll
  for (int d = 0; d < D_EMB; ++d) x[d] = ws[l * D_EMB + d];
#pragma unroll
  for (int o = 0; o < D_EMB; ++o) {
    float acc = ba[o];
#pragma unroll
    for (int d = 0; d < D_EMB; ++d) acc = fmaf(Wa[o * D_EMB + d], x[d], acc);
    hsAtt[l * D_EMB + o] = fmaxf(acc, 0.0f);
  }
}

__global__ void k_hvec(const float* __restrict__ compound, const float* __restrict__ Wa,
                       const float* __restrict__ ba, float* __restrict__ h) {
  int o = threadIdx.x;
  if (o < D_EMB) {
    float acc = ba[o];
#pragma unroll
    for (int d = 0; d < D_EMB; ++d) acc = fmaf(Wa[o * D_EMB + d], compound[d], acc);
    h[o] = fmaxf(acc, 0.0f);
  }
}

__global__ void k_protein(const float* __restrict__ hsAtt, const float* __restrict__ h,
                          float* __restrict__ protein) {
  __shared__ float red[256 * D_EMB];
  float hv[D_EMB], acc[D_EMB];
#pragma unroll
  for (int d = 0; d < D_EMB; ++d) { hv[d] = h[d]; acc[d] = 0.f; }
  for (int l = threadIdx.x; l < L_WORDS; l += 256) {
    float row[D_EMB], s = 0.f;
#pragma unroll
    for (int d = 0; d < D_EMB; ++d) { row[d] = hsAtt[l * D_EMB + d]; s = fmaf(hv[d], row[d], s); }
    s = tanhf(s);
#pragma unroll
    for (int d = 0; d < D_EMB; ++d) acc[d] = fmaf(s, row[d], acc[d]);
  }
#pragma unroll
  for (int d = 0; d < D_EMB; ++d) red[threadIdx.x * D_EMB + d] = acc[d];
  __syncthreads();
  if (threadIdx.x < D_EMB) {
    float s = 0.f;
    for (int t2 = 0; t2 < 256; ++t2) s += red[t2 * D_EMB + threadIdx.x];
    protein[threadIdx.x] = s / (float)L_WORDS;
  }
}

// ---------------- Output MLP ----------------
__global__ void k_final(const float* __restrict__ compound, const float* __restrict__ protein,
                        const float* __restrict__ Wo, const float* __restrict__ bo,
                        const float* __restrict__ Wi, const float* __restrict__ bi,
                        float* __restrict__ out) {
  __shared__ float cat[2 * 2 * D_EMB];
  int t = threadIdx.x;
  if (t < D_EMB) { cat[t] = compound[t]; cat[D_EMB + t] = protein[t]; }
  __syncthreads();
  for (int j = 0; j < 3; ++j) {
    float v = 0.f;
    if (t < 20) {
      v = bo[j * 20 + t];
      for (int k = 0; k < 20; ++k) v = fmaf(Wo[j * 400 + t * 20 + k], cat[k], v);
      v = fmaxf(v, 0.0f);
    }
    __syncthreads();
    if (t < 20) cat[t] = v;
    __syncthreads();
  }
  if (t < 2) {
    float v = bi[t];
    for (int k = 0; k < 20; ++k) v = fmaf(Wi[t * 20 + k], cat[k], v);
    out[t] = v;
  }
}

extern "C" void kernel_launch(void* const* d_in, const int* in_sizes, int n_in,
                              void* d_out, int out_size, void* d_ws, size_t ws_size,
                              hipStream_t stream) {
  const int*   fingerprints = (const int*)d_in[0];
  const float* adjacency    = (const float*)d_in[1];
  const int*   words        = (const int*)d_in[2];
  const float* emb_fp       = (const float*)d_in[3];
  const float* emb_word     = (const float*)d_in[4];
  const float* Wg_w         = (const float*)d_in[5];
  const float* Wg_b         = (const float*)d_in[6];
  const float* conv_w       = (const float*)d_in[7];
  const float* conv_b       = (const float*)d_in[8];
  const float* Wa_w         = (const float*)d_in[9];
  const float* Wa_b         = (const float*)d_in[10];
  const float* Wo_w         = (const float*)d_in[11];
  const float* Wo_b         = (const float*)d_in[12];
  const float* Wi_w         = (const float*)d_in[13];
  const float* Wi_b         = (const float*)d_in[14];
  float* out = (float*)d_out;

  // workspace partition (~2.1 MB of float scratch)
  float* wsf   = (float*)d_ws;
  float* bufA  = wsf;                         // N_ATOMS*DP
  float* bufB  = bufA + N_ATOMS * DP;         // N_ATOMS*DP
  float* hsP   = bufB + N_ATOMS * DP;         // N_ATOMS*DP
  float* wsA   = hsP + N_ATOMS * DP;          // L_WORDS*D_EMB
  float* wsB   = wsA + L_WORDS * D_EMB;       // L_WORDS*D_EMB
  float* hsAtt = wsB + L_WORDS * D_EMB;       // L_WORDS*D_EMB
  float* comp  = hsAtt + L_WORDS * D_EMB;     // 16
  float* hv    = comp + 16;                   // 16
  float* prot  = hv + 16;                     // 16

  // --- GNN ---
  k_gather_fp<<<N_ATOMS * DP / 256, 256, 0, stream>>>(fingerprints, emb_fp, bufA);
  float* cur = bufA; float* nxt = bufB;
  for (int i = 0; i < 3; ++i) {
    k_gnn_hidden<<<N_ATOMS / 256, 256, 0, stream>>>(cur, Wg_w + i * 100, Wg_b + i * 10, hsP);
    k_copy4<<<N_ATOMS * DP / 4 / 256, 256, 0, stream>>>((const float4*)cur, (float4*)nxt);
    k_adj_gemm<<<(N_ATOMS / ROWS) * KSPLIT, 256, 0, stream>>>(adjacency, hsP, nxt);
    float* tmp = cur; cur = nxt; nxt = tmp;
  }
  k_mean_rows<<<1, 256, 0, stream>>>(cur, comp);

  // --- Protein ---
  k_gather_word<<<L_WORDS * D_EMB / 256, 256, 0, stream>>>(words, emb_word, wsA);
  k_conv<<<L_WORDS * D_EMB / 256, 256, 0, stream>>>(wsA, conv_w, conv_b, 0, wsB);
  k_conv<<<L_WORDS * D_EMB / 256, 256, 0, stream>>>(wsB, conv_w, conv_b, 1, wsA);
  k_conv<<<L_WORDS * D_EMB / 256, 256, 0, stream>>>(wsA, conv_w, conv_b, 2, wsB);
  k_att<<<L_WORDS / 256, 256, 0, stream>>>(wsB, Wa_w, Wa_b, hsAtt);
  k_hvec<<<1, 32, 0, stream>>>(comp, Wa_w, Wa_b, hv);
  k_protein<<<1, 256, 0, stream>>>(hsAtt, hv, prot);

  // --- Output MLP ---
  k_final<<<1, 32, 0, stream>>>(comp, prot, Wo_w, Wo_b, Wi_w, Wi_b, out);
}